// QP_CasadiLayer_Query_24979529794071
// MI455X (gfx1250) — compile-verified
//
#include <hip/hip_runtime.h>

typedef float v2f __attribute__((ext_vector_type(2)));
typedef float v8f __attribute__((ext_vector_type(8)));
typedef int   v4i_ __attribute__((vector_size(16)));
typedef __attribute__((address_space(1))) v4i_* global_v4i_ptr;
typedef __attribute__((address_space(3))) v4i_* lds_v4i_ptr;

#define NDIM 64
#define NEQ  16
#define GSTR 65   // padded row stride for G (bank-conflict-free column walks)
#define WSTR 18   // W has 17 live columns (16 for A^T, 1 for q), pad to 18
#define SSTR 17

#define TAILF (NDIM + NEQ*NDIM + NEQ)   // q(64) + A(1024) + b(16) = 1104 floats
#define TAIL4 (TAILF / 4)               // 276 float4 chunks

#if __has_builtin(__builtin_amdgcn_global_load_async_to_lds_b128) && \
    __has_builtin(__builtin_amdgcn_s_wait_asynccnt)
#define USE_ASYNC_LDS 1
#endif

// One wave32 per batch element. Schur-complement KKT solve:
//   L L^T = sym(Q);  W = L^-1 A^T;  z = L^-1 q;
//   S = W^T W  (WMMA f32 16x16x4, K=64);  lam = S^-1 (b - W^T z);
//   x = -(L^T)^-1 (z + W lam)
__global__ __launch_bounds__(32)
void qp_kkt_solve(const float* __restrict__ lp, float* __restrict__ out)
{
    const int b    = blockIdx.x;
    const int lane = threadIdx.x;

    const int per = NDIM*NDIM + TAILF;                 // 5200 floats per item
    const float* base = lp + (size_t)b * per;
    const float* Qp   = base;
    const float* tail = base + NDIM*NDIM;              // q | A | b contiguous

    __shared__ __attribute__((aligned(16))) float G[NDIM*GSTR]; // sym(Q) -> L (lower)
    __shared__ __attribute__((aligned(16))) float AQ[TAILF];    // staged q|A|b
    __shared__ float W[NDIM*WSTR];   // cols 0..15: L^-1 A^T ; col 16: z = L^-1 q
    __shared__ float S[NEQ*SSTR];    // Schur complement, then its Cholesky factor
    __shared__ float lam[NEQ];
    __shared__ float xv[NDIM];       // v = z + W*lam, then back-solve in place

    // ---- kick off async staging of q|A|b into LDS; hidden under the Cholesky ----
#ifdef USE_ASYNC_LDS
    for (int t = lane; t < TAIL4; t += 32)
        __builtin_amdgcn_global_load_async_to_lds_b128(
            (global_v4i_ptr)(tail + 4*t),
            (lds_v4i_ptr)(AQ + 4*t),
            0, 0);
#else
    for (int t = lane; t < TAIL4; t += 32)
        ((float4*)AQ)[t] = ((const float4*)tail)[t];
#endif

    // ---- coalesced B128 load of Q into LDS ----
    for (int t = lane; t < (NDIM*NDIM)/4; t += 32) {
        float4 v4 = ((const float4*)Qp)[t];
        const int r = t >> 4;             // row
        const int c = (t & 15) << 2;      // col
        G[r*GSTR + c + 0] = v4.x;
        G[r*GSTR + c + 1] = v4.y;
        G[r*GSTR + c + 2] = v4.z;
        G[r*GSTR + c + 3] = v4.w;
    }
    __syncthreads();

    // ---- symmetrize lower triangle in place (Cholesky reads lower only) ----
    for (int r = lane; r < NDIM; r += 32)
        for (int j = 0; j <= r; ++j)
            G[r*GSTR + j] = 0.5f * (G[r*GSTR + j] + G[j*GSTR + r]);
    __syncthreads();

    // ---- Cholesky of G (lower, in place) ----
    for (int k = 0; k < NDIM; ++k) {
        float dk  = sqrtf(G[k*GSTR + k]);   // all lanes read (before lane0 store)
        float inv = 1.0f / dk;
        if (lane == 0) G[k*GSTR + k] = dk;
        for (int i = k + 1 + lane; i < NDIM; i += 32)
            G[i*GSTR + k] *= inv;
        __syncthreads();
        for (int i = lane; i < NDIM; i += 32) {
            if (i <= k) continue;
            float lik = G[i*GSTR + k];
            for (int j = k + 1; j <= i; ++j)
                G[i*GSTR + j] -= lik * G[j*GSTR + k];
        }
        __syncthreads();
    }

    // ---- staged tail must be resident before the solves ----
#ifdef USE_ASYNC_LDS
    __builtin_amdgcn_s_wait_asynccnt(0);
#endif
    __syncthreads();

    // ---- forward solve: 17 RHS columns (A^T cols + q), lanes 0..16 each own one ----
    // AQ layout: q = AQ[0..63], A = AQ[64 .. 64+1023] (row-major 16x64), b = AQ[1088..1103]
    if (lane < 17) {
        for (int i = 0; i < NDIM; ++i) {
            float s = (lane < NEQ) ? AQ[NDIM + lane*NDIM + i]   // A^T[i][c] = A[c][i]
                                   : AQ[i];                     // q
            for (int j = 0; j < i; ++j)
                s -= G[i*GSTR + j] * W[j*WSTR + lane];
            W[i*WSTR + lane] = s / G[i*GSTR + i];
        }
    }
    __syncthreads();

    // ---- S = W^T W via V_WMMA_F32_16X16X4_F32, K accumulated over 64 in steps of 4 ----
    // A' = W^T (16x4 slab): lane m=l%16 holds rows M=m; VGPR pair covers K = k0+2*(l/16)+{0,1}
    // B' = W   (4x16 slab): lane n=l%16 holds cols N=n; same K split -> identical registers.
    {
        const int m  = lane & 15;
        const int kh = lane >> 4;           // 0: K offsets {0,1}; 1: {2,3}
        v8f acc = {};
        for (int kk = 0; kk < 16; ++kk) {
            const int k0 = 4*kk + 2*kh;
            v2f a;
            a.x = W[(k0 + 0) * WSTR + m];
            a.y = W[(k0 + 1) * WSTR + m];
            acc = __builtin_amdgcn_wmma_f32_16x16x4_f32(
                false, a, false, a, (short)0, acc, false, false);
        }
        // C/D layout: lane l holds S[M][N] with N=l%16, M = vreg + 8*(l/16)
        for (int v = 0; v < 8; ++v)
            S[(v + 8*kh) * SSTR + m] = acc[v];
    }
    __syncthreads();

    // ---- rhs for lambda: b - W^T z ----
    if (lane < NEQ) {
        float s = AQ[NDIM + NEQ*NDIM + lane];            // b
        for (int k = 0; k < NDIM; ++k)
            s -= W[k*WSTR + lane] * W[k*WSTR + 16];
        lam[lane] = s;
    }
    __syncthreads();

    // ---- Cholesky of S (16x16), lanes 0..15 ----
    for (int k = 0; k < NEQ; ++k) {
        float dk  = sqrtf(S[k*SSTR + k]);
        float inv = 1.0f / dk;
        if (lane == 0) S[k*SSTR + k] = dk;
        if (lane > k && lane < NEQ) S[lane*SSTR + k] *= inv;
        __syncthreads();
        if (lane > k && lane < NEQ) {
            float lik = S[lane*SSTR + k];
            for (int j = k + 1; j <= lane; ++j)
                S[lane*SSTR + j] -= lik * S[j*SSTR + k];
        }
        __syncthreads();
    }

    // ---- solve S lam (tiny: serial on lane 0) ----
    if (lane == 0) {
        for (int i = 0; i < NEQ; ++i) {
            float s = lam[i];
            for (int j = 0; j < i; ++j) s -= S[i*SSTR + j] * lam[j];
            lam[i] = s / S[i*SSTR + i];
        }
        for (int i = NEQ - 1; i >= 0; --i) {
            float s = lam[i];
            for (int j = i + 1; j < NEQ; ++j) s -= S[j*SSTR + i] * lam[j];
            lam[i] = s / S[i*SSTR + i];
        }
    }
    __syncthreads();

    // ---- v = z + W lam ----
    for (int i = lane; i < NDIM; i += 32) {
        float s = W[i*WSTR + 16];
        for (int mth = 0; mth < NEQ; ++mth)
            s += W[i*WSTR + mth] * lam[mth];
        xv[i] = s;
    }
    __syncthreads();

    // ---- x = (L^T)^-1 v, cooperative back-substitution (shuffle reduction) ----
    for (int i = NDIM - 1; i >= 0; --i) {
        float partial = 0.0f;
        for (int j = i + 1 + lane; j < NDIM; j += 32)
            partial += G[j*GSTR + i] * xv[j];
        for (int off = 16; off > 0; off >>= 1)
            partial += __shfl_down(partial, off, 32);
        if (lane == 0)
            xv[i] = (xv[i] - partial) / G[i*GSTR + i];
        __syncthreads();
    }

    // ---- output sol[:64] = -x ----
    for (int i = lane; i < NDIM; i += 32)
        out[(size_t)b * NDIM + i] = -xv[i];
}

extern "C" void kernel_launch(void* const* d_in, const int* in_sizes, int n_in,
                              void* d_out, int out_size, void* d_ws, size_t ws_size,
                              hipStream_t stream) {
    const float* lp  = (const float*)d_in[0];
    float*       out = (float*)d_out;
    const int per = NDIM*NDIM + NDIM + NEQ*NDIM + NEQ; // 5200
    const int batch = in_sizes[0] / per;               // 8192
    qp_kkt_solve<<<batch, 32, 0, stream>>>(lp, out);
}